// Model_66778151518719
// MI455X (gfx1250) — compile-verified
//
#include <hip/hip_runtime.h>

typedef float v2f __attribute__((ext_vector_type(2)));
typedef float v8f __attribute__((ext_vector_type(8)));

#define NNODES 5000
#define KVIRT  30
#define BATCH  16
#define DIN    12
#define HDIM   64
#define MTOTAL (BATCH * NNODES)   // 80000, divisible by 128

// ---------------------------------------------------------------------------
// Kernel 1: per-node diagonal scale  s[i] = adj[i,i] / deg[i]
//   deg[i] = sum_j adj[i,j] + sum_c adjk1[c,i]
// One 256-thread block per node; LDS tree reduction.
// ---------------------------------------------------------------------------
__global__ void __launch_bounds__(256)
scale_kernel(const float* __restrict__ adj,
             const float* __restrict__ adjk1,
             float* __restrict__ s) {
    __shared__ float red[256];
    const int i   = blockIdx.x;
    const int tid = threadIdx.x;

    const float* row = adj + (size_t)i * NNODES;
    float partial = 0.f;
    for (int j = tid; j < NNODES; j += 256) partial += row[j];
    for (int c = tid; c < KVIRT;  c += 256) partial += adjk1[c * NNODES + i];

    red[tid] = partial;
    __syncthreads();
    for (int off = 128; off > 0; off >>= 1) {
        if (tid < off) red[tid] += red[tid + off];
        __syncthreads();
    }
    if (tid == 0) s[i] = row[i] / red[0];
}

// ---------------------------------------------------------------------------
// Kernel 2: fused GCN layer with diagonal adjacency:
//   out[row,:] = tanh(s[row%N]*(in@W)[row,:] + bias)
//             * sigmoid(s[row%N]*(in@Wp)[row,:] + biasp)
//
// Each wave computes a 16x64 output tile with V_WMMA_F32_16X16X4_F32.
// KDIM is a template parameter so the K loop fully unrolls.
//
// Weights are staged in LDS *pre-swizzled* so each lane's B operand pair
// (W[k][n], W[k+1][n]) is contiguous and 8B aligned -> single ds_load_b64
// directly into an even-aligned VGPR pair (no pack movs):
//     sW[((k>>1)*64 + n)*2 + (k&1)] = W[k*64 + n]
//
// Operand layouts per ISA 7.12.2 (f32, wave32):
//   A (16x4):  lane l: M=l&15, VGPR0 K=k0+2*(l>>4), VGPR1 K+1
//   B (4x16):  lane l: N=l&15, VGPR0 K=k0+2*(l>>4), VGPR1 K+1
//   C/D (16x16): VGPR r, lane l -> M=r+8*(l>>4), N=l&15
// ---------------------------------------------------------------------------
template <int KDIM>
__global__ void __launch_bounds__(256)
gcn_layer(const float* __restrict__ in,
          const float* __restrict__ W,  const float* __restrict__ bias,
          const float* __restrict__ Wp, const float* __restrict__ biasp,
          const float* __restrict__ s,
          float* __restrict__ out, int mtotal) {
    __shared__ float sW [HDIM * HDIM];
    __shared__ float sWp[HDIM * HDIM];

    const int tid = threadIdx.x;
    // Cooperative swizzled staging of both weight matrices.
    for (int idx = tid; idx < KDIM * HDIM; idx += 256) {
        const int k = idx >> 6;          // idx / HDIM
        const int n = idx & (HDIM - 1);  // idx % HDIM
        const int pos = (((k >> 1) * HDIM + n) << 1) | (k & 1);
        sW[pos]  = W[idx];
        sWp[pos] = Wp[idx];
    }
    __syncthreads();

    const int wave    = tid >> 5;        // wave32
    const int lane    = tid & 31;
    const int rowBase = blockIdx.x * 128 + wave * 16;
    if (rowBase >= mtotal) return;       // wave-uniform: EXEC stays all-ones

    const int m    = lane & 15;
    const int half = lane >> 4;

    const v8f zero = {0.f, 0.f, 0.f, 0.f, 0.f, 0.f, 0.f, 0.f};
    v8f acc[4], accp[4];
#pragma unroll
    for (int t = 0; t < 4; ++t) { acc[t] = zero; accp[t] = zero; }

    const float* arow = in + (size_t)(rowBase + m) * KDIM;
#pragma unroll
    for (int k0 = 0; k0 < KDIM; k0 += 4) {
        const int kb = k0 + 2 * half;                  // even
        const v2f a = *(const v2f*)&arow[kb];          // global_load_b64
#pragma unroll
        for (int t = 0; t < 4; ++t) {
            const int n   = t * 16 + m;
            const int pos = (((kb >> 1) * HDIM + n) << 1);
            const v2f b  = *(const v2f*)&sW[pos];      // ds_load_b64
            acc[t] = __builtin_amdgcn_wmma_f32_16x16x4_f32(
                false, a, false, b, (short)0, acc[t], false, false);
            const v2f bp = *(const v2f*)&sWp[pos];     // ds_load_b64
            accp[t] = __builtin_amdgcn_wmma_f32_16x16x4_f32(
                false, a, false, bp, (short)0, accp[t], false, false);
        }
    }

    // Epilogue: node index via one wave-uniform mod + cheap wrap per row.
    const int nodeBase = rowBase % NNODES;             // scalar (wave-uniform)
#pragma unroll
    for (int r = 0; r < 8; ++r) {
        const int rr   = r + 8 * half;
        const int row  = rowBase + rr;
        int node = nodeBase + rr;
        if (node >= NNODES) node -= NNODES;
        const float sc = s[node];
#pragma unroll
        for (int t = 0; t < 4; ++t) {
            const int col   = t * 16 + m;
            const float h   = sc * acc[t][r]  + bias[col];
            const float hp  = sc * accp[t][r] + biasp[col];
            const float th  = 1.0f - 2.0f / (__expf(2.0f * h) + 1.0f);   // tanh
            const float sig = 1.0f / (1.0f + __expf(-hp));               // sigmoid
            out[(size_t)row * HDIM + col] = th * sig;
        }
    }
}

// ---------------------------------------------------------------------------
// Launcher
// ---------------------------------------------------------------------------
extern "C" void kernel_launch(void* const* d_in, const int* in_sizes, int n_in,
                              void* d_out, int out_size, void* d_ws, size_t ws_size,
                              hipStream_t stream) {
    (void)in_sizes; (void)n_in; (void)out_size; (void)ws_size;

    const float* x     = (const float*)d_in[0];   // (B, N, D)
    const float* adj   = (const float*)d_in[1];   // (N, N)
    const float* adjk1 = (const float*)d_in[2];   // (K, N)
    const float* W1    = (const float*)d_in[3];
    const float* b1    = (const float*)d_in[4];
    const float* W11   = (const float*)d_in[5];
    const float* b11   = (const float*)d_in[6];
    const float* W2    = (const float*)d_in[7];
    const float* b2    = (const float*)d_in[8];
    const float* W22   = (const float*)d_in[9];
    const float* b22   = (const float*)d_in[10];
    const float* W3    = (const float*)d_in[11];
    const float* b3    = (const float*)d_in[12];
    const float* W33   = (const float*)d_in[13];
    const float* b33   = (const float*)d_in[14];

    float* out  = (float*)d_out;
    float* ws   = (float*)d_ws;
    float* sArr = ws;                                  // N floats
    float* buf0 = ws + 8192;                           // [80000, 64]
    float* buf1 = buf0 + (size_t)MTOTAL * HDIM;        // [80000, 64]

    const dim3 blk(256);
    scale_kernel<<<dim3(NNODES), blk, 0, stream>>>(adj, adjk1, sArr);

    const dim3 grid((MTOTAL + 127) / 128);             // 625
    gcn_layer<DIN> <<<grid, blk, 0, stream>>>(x,    W1, b1, W11, b11, sArr, buf0, MTOTAL);
    gcn_layer<HDIM><<<grid, blk, 0, stream>>>(buf0, W2, b2, W22, b22, sArr, buf1, MTOTAL);
    gcn_layer<HDIM><<<grid, blk, 0, stream>>>(buf1, W3, b3, W33, b33, sArr, out,  MTOTAL);
}